// MultiBodyLayer_6734508720710
// MI455X (gfx1250) — compile-verified
//
#include <hip/hip_runtime.h>

// ---------------------------------------------------------------------------
// Fused multi-body invariant-feature + channel-mixing GEMM for MI455X (gfx1250)
// f32 end-to-end via V_WMMA_F32_16X16X4_F32 (wave32 WMMA, f32 A/B/C/D).
//
// Assumed wave32 fragment layouts (CDNA5 ISA 7.12.2):
//   A (16x4 f32):  lane = m + 16*(k>>1), vgpr v = k&1        -> v2f
//   B (4x16 f32):  lane = n + 16*(k>>1), vgpr v = k&1        -> v2f (mirror of A)
//   C/D (16x16):   vgpr p: row m = p + 8*(lane>>4), col n = lane&15 -> v8f
//
// LDS feature buffer is [comp][row][ch] with row stride 132 (CCH+4) so each
// lane's K-pair is one 8B-aligned ds_load_b64 and the 32 lanes of a fragment
// read hit 64 distinct banks.  Weights are repacked once into d_ws as
// [k_pair][col][2] so each B fragment is one coalesced global_load_b64.
// ---------------------------------------------------------------------------

typedef __attribute__((ext_vector_type(2))) float v2f;
typedef __attribute__((ext_vector_type(8))) float v8f;

#define NROWS    10000
#define TILE_M   16
#define CCH      128
#define OUTD     128
#define NTHREADS 256
#define ROWP     (CCH + 4)          // padded row stride (dwords) in s_feat
#define FCOMP    (TILE_M * ROWP)    // per-component stride in s_feat

__device__ __forceinline__ v8f wmma4(v2f a, v2f b, v8f c) {
  // 8-arg form: (neg_a, A, neg_b, B, c_mod, C, reuse_a, reuse_b)
  return __builtin_amdgcn_wmma_f32_16x16x4_f32(false, a, false, b, (short)0, c,
                                               false, false);
}

// B fragment: K-pair (c0, c0+1) at column `col`.
template <bool PACKED>
__device__ __forceinline__ v2f load_b(const float* __restrict__ Wt, int c0,
                                      int col) {
  if (PACKED) {
    return *(const v2f*)&Wt[(((size_t)(c0 >> 1) * OUTD + col)) << 1];
  } else {
    v2f b = {Wt[(size_t)c0 * OUTD + col], Wt[(size_t)(c0 + 1) * OUTD + col]};
    return b;
  }
}

template <bool PACKED>
__global__ __launch_bounds__(NTHREADS, 1) void multibody_fused(
    const float* __restrict__ x0, const float* __restrict__ x1,
    const float* __restrict__ x2, const float* __restrict__ W0,
    const float* __restrict__ b0, const float* __restrict__ W1,
    const float* __restrict__ W2, float* __restrict__ y0,
    float* __restrict__ y1, float* __restrict__ y2) {
  extern __shared__ float lds[];
  float* s_in   = lds;                      // [16*128][13] raw a,v,M  (104 KB)
  float* s_feat = lds + TILE_M * CCH * 13;  // [comp<=9][16][132]      (~74 KB)

  const int tid  = threadIdx.x;
  const int lane = tid & 31;
  const int wv   = tid >> 5;          // 8 waves, one 16-col tile each
  const int m    = lane & 15;
  const int kh   = lane >> 4;         // which K-half of the fragment
  const int col  = wv * 16 + m;       // this lane's output column
  const int row0 = blockIdx.x * TILE_M;

  // ---- stage raw inputs for 16 rows into LDS (streamed: non-temporal) ----
  for (int p = tid; p < TILE_M * CCH; p += NTHREADS) {
    const int r = p >> 7, ch = p & (CCH - 1);
    const size_t bc = (size_t)(row0 + r) * CCH + ch;
    float* dst = &s_in[p * 13];
    dst[0] = __builtin_nontemporal_load(&x0[bc]);
    dst[1] = __builtin_nontemporal_load(&x1[bc * 3 + 0]);
    dst[2] = __builtin_nontemporal_load(&x1[bc * 3 + 1]);
    dst[3] = __builtin_nontemporal_load(&x1[bc * 3 + 2]);
#pragma unroll
    for (int j = 0; j < 9; ++j)
      dst[4 + j] = __builtin_nontemporal_load(&x2[bc * 9 + j]);
  }
  __syncthreads();

  const v8f vzero = {0.f, 0.f, 0.f, 0.f, 0.f, 0.f, 0.f, 0.f};

  //========================== w = 0 : y0 = cat0 @ W0 + b0 ====================
  {
    v8f acc = vzero;
    for (int t = 0; t < 12; ++t) {
      for (int p = tid; p < TILE_M * CCH; p += NTHREADS) {
        const float* in = &s_in[p * 13];
        const float a = in[0], u0 = in[1], u1 = in[2], u2 = in[3];
        const float* M = in + 4;
        const float d11 = u0 * u0 + u1 * u1 + u2 * u2;
        float f22 = 0.f;
#pragma unroll
        for (int j = 0; j < 9; ++j) f22 += M[j] * M[j];
        float val;
        switch (t) {
          case 0: val = a; break;
          case 1: val = a * a; break;
          case 2: val = d11; break;
          case 3: val = f22; break;
          case 4: val = a * a * a; break;
          case 5: case 7: val = a * d11; break;
          case 6: case 9: val = a * f22; break;
          case 8: case 10: {  // v^T M v
            const float mv0 = M[0] * u0 + M[1] * u1 + M[2] * u2;
            const float mv1 = M[3] * u0 + M[4] * u1 + M[5] * u2;
            const float mv2 = M[6] * u0 + M[7] * u1 + M[8] * u2;
            val = mv0 * u0 + mv1 * u1 + mv2 * u2;
          } break;
          default: {  // t=11 : sum_ij (M M^T)_ij M_ij
            float s = 0.f;
#pragma unroll
            for (int i = 0; i < 3; ++i)
#pragma unroll
              for (int j = 0; j < 3; ++j) {
                const float g = M[i * 3 + 0] * M[j * 3 + 0] +
                                M[i * 3 + 1] * M[j * 3 + 1] +
                                M[i * 3 + 2] * M[j * 3 + 2];
                s += g * M[i * 3 + j];
              }
            val = s;
          } break;
        }
        s_feat[(p >> 7) * ROWP + (p & (CCH - 1))] = val;  // [row][ch]
      }
      __syncthreads();

      const float* Wt = W0 + (size_t)t * CCH * OUTD;
      if (t + 1 < 12) __builtin_prefetch(Wt + CCH * OUTD + tid * 64, 0, 3);
#pragma unroll 4
      for (int ks = 0; ks < CCH / 4; ++ks) {
        const int c0 = 4 * ks + 2 * kh;
        v2f bf = load_b<PACKED>(Wt, c0, col);
        v2f af = *(const v2f*)&s_feat[m * ROWP + c0];
        acc = wmma4(af, bf, acc);
      }
      __syncthreads();
    }
    const float bias = b0[col];
#pragma unroll
    for (int p8 = 0; p8 < 8; ++p8) {
      const int br = p8 + 8 * kh;
      __builtin_nontemporal_store(acc[p8] + bias,
                                  &y0[(size_t)(row0 + br) * OUTD + col]);
    }
  }

  //========================== w = 1 : y1[b,d,i] ==============================
  {
    v8f acc[3] = {vzero, vzero, vzero};
    for (int t = 0; t < 8; ++t) {
      for (int p = tid; p < TILE_M * CCH; p += NTHREADS) {
        const float* in = &s_in[p * 13];
        const float a = in[0];
        const float u[3] = {in[1], in[2], in[3]};
        const float* M = in + 4;
        float mv[3];
#pragma unroll
        for (int i = 0; i < 3; ++i)
          mv[i] = M[i * 3 + 0] * u[0] + M[i * 3 + 1] * u[1] + M[i * 3 + 2] * u[2];
        const float d11 = u[0] * u[0] + u[1] * u[1] + u[2] * u[2];
        float f22 = 0.f;
#pragma unroll
        for (int j = 0; j < 9; ++j) f22 += M[j] * M[j];
        float val[3];
#pragma unroll
        for (int i = 0; i < 3; ++i) {
          switch (t) {
            case 0: val[i] = u[i]; break;
            case 1: val[i] = a * u[i]; break;
            case 2: val[i] = mv[i]; break;
            case 3: val[i] = a * a * u[i]; break;
            case 4: val[i] = d11 * u[i]; break;
            case 5: val[i] = f22 * u[i]; break;
            case 6: val[i] = a * mv[i]; break;
            default:  // t=7 : M(Mv)
              val[i] = M[i * 3 + 0] * mv[0] + M[i * 3 + 1] * mv[1] +
                       M[i * 3 + 2] * mv[2];
              break;
          }
        }
        const int ch = p & (CCH - 1), r = p >> 7;
#pragma unroll
        for (int i = 0; i < 3; ++i) s_feat[i * FCOMP + r * ROWP + ch] = val[i];
      }
      __syncthreads();

      const float* Wt = W1 + (size_t)t * CCH * OUTD;
      if (t + 1 < 8) __builtin_prefetch(Wt + CCH * OUTD + tid * 64, 0, 3);
#pragma unroll 2
      for (int ks = 0; ks < CCH / 4; ++ks) {
        const int c0 = 4 * ks + 2 * kh;
        v2f bf = load_b<PACKED>(Wt, c0, col);
#pragma unroll
        for (int i = 0; i < 3; ++i) {
          v2f af = *(const v2f*)&s_feat[i * FCOMP + m * ROWP + c0];
          acc[i] = wmma4(af, bf, acc[i]);
        }
      }
      __syncthreads();
    }
#pragma unroll
    for (int i = 0; i < 3; ++i)
#pragma unroll
      for (int p8 = 0; p8 < 8; ++p8) {
        const int br = p8 + 8 * kh;
        __builtin_nontemporal_store(
            acc[i][p8], &y1[((size_t)(row0 + br) * OUTD + col) * 3 + i]);
      }
  }

  //========================== w = 2 : y2[b,d,i,j] ============================
  {
    v8f acc[9];
#pragma unroll
    for (int q = 0; q < 9; ++q) acc[q] = vzero;
    for (int t = 0; t < 12; ++t) {
      for (int p = tid; p < TILE_M * CCH; p += NTHREADS) {
        const float* in = &s_in[p * 13];
        const float a = in[0];
        const float u[3] = {in[1], in[2], in[3]};
        const float* M = in + 4;
        float mv[3];
#pragma unroll
        for (int i = 0; i < 3; ++i)
          mv[i] = M[i * 3 + 0] * u[0] + M[i * 3 + 1] * u[1] + M[i * 3 + 2] * u[2];
        float G[9];  // M M^T
#pragma unroll
        for (int i = 0; i < 3; ++i)
#pragma unroll
          for (int j = 0; j < 3; ++j)
            G[i * 3 + j] = M[i * 3 + 0] * M[j * 3 + 0] +
                           M[i * 3 + 1] * M[j * 3 + 1] +
                           M[i * 3 + 2] * M[j * 3 + 2];
        const float d11 = u[0] * u[0] + u[1] * u[1] + u[2] * u[2];
        float f22 = 0.f;
#pragma unroll
        for (int j = 0; j < 9; ++j) f22 += M[j] * M[j];
        float val[9];
#pragma unroll
        for (int i = 0; i < 3; ++i)
#pragma unroll
          for (int j = 0; j < 3; ++j) {
            const int q = i * 3 + j;
            switch (t) {
              case 0: val[q] = M[q]; break;
              case 1: val[q] = a * M[q]; break;
              case 2: val[q] = u[i] * u[j]; break;
              case 3: val[q] = G[q]; break;
              case 4: val[q] = a * a * M[q]; break;
              case 5: val[q] = d11 * M[q]; break;
              case 6: val[q] = f22 * M[q]; break;
              case 7: val[q] = a * u[i] * u[j]; break;
              case 8: val[q] = mv[i] * u[j]; break;
              case 9: val[q] = a * G[q]; break;
              case 10: val[q] = u[i] * mv[j]; break;
              default:  // t=11 : (M M^T) M^T
                val[q] = G[i * 3 + 0] * M[j * 3 + 0] +
                         G[i * 3 + 1] * M[j * 3 + 1] +
                         G[i * 3 + 2] * M[j * 3 + 2];
                break;
            }
          }
        const int ch = p & (CCH - 1), r = p >> 7;
#pragma unroll
        for (int q = 0; q < 9; ++q) s_feat[q * FCOMP + r * ROWP + ch] = val[q];
      }
      __syncthreads();

      const float* Wt = W2 + (size_t)t * CCH * OUTD;
      if (t + 1 < 12) __builtin_prefetch(Wt + CCH * OUTD + tid * 64, 0, 3);
      for (int ks = 0; ks < CCH / 4; ++ks) {
        const int c0 = 4 * ks + 2 * kh;
        v2f bf = load_b<PACKED>(Wt, c0, col);
#pragma unroll
        for (int q = 0; q < 9; ++q) {
          v2f af = *(const v2f*)&s_feat[q * FCOMP + m * ROWP + c0];
          acc[q] = wmma4(af, bf, acc[q]);
        }
      }
      __syncthreads();
    }
#pragma unroll
    for (int q = 0; q < 9; ++q)
#pragma unroll
      for (int p8 = 0; p8 < 8; ++p8) {
        const int br = p8 + 8 * kh;
        __builtin_nontemporal_store(
            acc[q][p8], &y2[((size_t)(row0 + br) * OUTD + col) * 9 + q]);
      }
  }
}

// Repack W (rows x 128, row-major) -> [row_pair][col][2] so a B fragment is a
// single coalesced global_load_b64.
__global__ __launch_bounds__(256) void repack_w(const float* __restrict__ src,
                                               float* __restrict__ dst,
                                               int nelem) {
  const int idx = blockIdx.x * 256 + threadIdx.x;
  if (idx < nelem) {
    const int c = idx >> 7, d = idx & 127;
    dst[((((size_t)(c >> 1)) * OUTD + d) << 1) + (c & 1)] = src[idx];
  }
}

extern "C" void kernel_launch(void* const* d_in, const int* in_sizes, int n_in,
                              void* d_out, int out_size, void* d_ws,
                              size_t ws_size, hipStream_t stream) {
  const float* x0 = (const float*)d_in[0];
  const float* x1 = (const float*)d_in[1];
  const float* x2 = (const float*)d_in[2];
  const float* W0 = (const float*)d_in[3];
  const float* b0 = (const float*)d_in[4];
  const float* W1 = (const float*)d_in[5];
  const float* W2 = (const float*)d_in[6];

  float* y0 = (float*)d_out;
  float* y1 = y0 + (size_t)NROWS * OUTD;      // (N,128,3) after (N,128)
  float* y2 = y1 + (size_t)NROWS * OUTD * 3;  // (N,128,3,3)

  constexpr size_t LDS_BYTES =
      (size_t)(TILE_M * CCH * 13 + 9 * FCOMP) * sizeof(float);  // ~178 KB

  const size_t n0 = (size_t)12 * CCH * OUTD;  // W0 elems
  const size_t n1 = (size_t)8 * CCH * OUTD;   // W1 elems
  const size_t n2 = (size_t)12 * CCH * OUTD;  // W2 elems
  const bool packed = ws_size >= (n0 + n1 + n2) * sizeof(float);

  if (packed) {
    float* w0p = (float*)d_ws;
    float* w1p = w0p + n0;
    float* w2p = w1p + n1;
    repack_w<<<((int)n0 + 255) / 256, 256, 0, stream>>>(W0, w0p, (int)n0);
    repack_w<<<((int)n1 + 255) / 256, 256, 0, stream>>>(W1, w1p, (int)n1);
    repack_w<<<((int)n2 + 255) / 256, 256, 0, stream>>>(W2, w2p, (int)n2);
    (void)hipFuncSetAttribute((const void*)multibody_fused<true>,
                              hipFuncAttributeMaxDynamicSharedMemorySize,
                              (int)LDS_BYTES);
    multibody_fused<true><<<NROWS / TILE_M, NTHREADS, LDS_BYTES, stream>>>(
        x0, x1, x2, w0p, b0, w1p, w2p, y0, y1, y2);
  } else {
    (void)hipFuncSetAttribute((const void*)multibody_fused<false>,
                              hipFuncAttributeMaxDynamicSharedMemorySize,
                              (int)LDS_BYTES);
    multibody_fused<false><<<NROWS / TILE_M, NTHREADS, LDS_BYTES, stream>>>(
        x0, x1, x2, W0, b0, W1, W2, y0, y1, y2);
  }
}